// Op1_73495480369226
// MI455X (gfx1250) — compile-verified
//
#include <hip/hip_runtime.h>
#include <stdint.h>

// ---------------------------------------------------------------------------
// COO SpMM scatter:  out[r,:] += ws[idx] * vals[e] * x[cols[e],:]
//   x: [50000,64] f32   edges: 800000   out: [50000,64] f32
// Memory-bound; x (12.8MB) and out (12.8MB) are L2-resident (192MB L2).
// CDNA5 path used: global_load_async_to_lds_b32 + s_wait_asynccnt to stage
// edge triples in LDS; wave32-aware 16-lane-per-edge feature mapping;
// hardware global_atomic_add_f32 scatter.
// ---------------------------------------------------------------------------

#define EDGES_PER_BLOCK 256
#define THREADS 256

__device__ __forceinline__ void async_load_dword_to_lds(unsigned lds_byte_off,
                                                        const void* gaddr) {
    // GV mode: VDST = per-lane LDS byte address, VADDR = 64-bit global addr.
    asm volatile("global_load_async_to_lds_b32 %0, %1, off"
                 :: "v"(lds_byte_off), "v"(gaddr)
                 : "memory");
}

__device__ __forceinline__ void wait_async_zero() {
    asm volatile("s_wait_asynccnt 0" ::: "memory");
}

__global__ void __launch_bounds__(THREADS)
spmm_zero_kernel(float4* __restrict__ out4, int n4) {
    int i = blockIdx.x * blockDim.x + threadIdx.x;
    if (i < n4) out4[i] = make_float4(0.f, 0.f, 0.f, 0.f);
}

__global__ void __launch_bounds__(THREADS)
spmm_scatter_kernel(const float4* __restrict__ x4,      // [N, 16] float4 rows
                    const int*    __restrict__ rows,
                    const int*    __restrict__ cols,
                    const float*  __restrict__ vals,
                    const float*  __restrict__ ws,
                    const int*    __restrict__ idx,
                    float*        __restrict__ out,
                    int E) {
    __shared__ int   s_rows[EDGES_PER_BLOCK];
    __shared__ int   s_cols[EDGES_PER_BLOCK];
    __shared__ float s_vals[EDGES_PER_BLOCK];

    const int tid = threadIdx.x;
    const int e0  = blockIdx.x * EDGES_PER_BLOCK;
    const int e   = e0 + tid;

    const float w = ws[idx[0]];

    // Stage this block's edge triples into LDS via the CDNA5 async path.
    if (e < E) {
        async_load_dword_to_lds((unsigned)(uintptr_t)&s_rows[tid], rows + e);
        async_load_dword_to_lds((unsigned)(uintptr_t)&s_cols[tid], cols + e);
        async_load_dword_to_lds((unsigned)(uintptr_t)&s_vals[tid], vals + e);
    }
    wait_async_zero();      // wave's own async loads complete (LDS written)
    __syncthreads();        // all 8 waves' LDS writes visible

    const int n_local   = (E - e0 < EDGES_PER_BLOCK) ? (E - e0) : EDGES_PER_BLOCK;
    const int q         = tid & 15;   // feature quad: 4 floats -> 16 quads = 64 feats
    const int lane_grp  = tid >> 4;   // 16 edges in flight per iteration

    #pragma unroll 4
    for (int sub = 0; sub < EDGES_PER_BLOCK / 16; ++sub) {
        const int el = lane_grp + sub * 16;
        if (el < n_local) {
            // 16 lanes broadcast-read the same edge triple from LDS.
            const int   r = s_rows[el];
            const int   c = s_cols[el];
            const float v = s_vals[el] * w;

            // Coalesced 256B row gather: lanes q=0..15 fetch x[c, 4q..4q+3].
            const float4 xv = x4[(size_t)c * 16 + q];

            float* o = out + (size_t)r * 64 + q * 4;
            atomicAdd(o + 0, v * xv.x);
            atomicAdd(o + 1, v * xv.y);
            atomicAdd(o + 2, v * xv.z);
            atomicAdd(o + 3, v * xv.w);
        }
    }
}

extern "C" void kernel_launch(void* const* d_in, const int* in_sizes, int n_in,
                              void* d_out, int out_size, void* d_ws, size_t ws_size,
                              hipStream_t stream) {
    const float* x    = (const float*)d_in[0];
    const int*   rows = (const int*)  d_in[1];
    const int*   cols = (const int*)  d_in[2];
    const float* vals = (const float*)d_in[3];
    const float* ws   = (const float*)d_in[4];
    const int*   idx  = (const int*)  d_in[5];
    float*       out  = (float*)d_out;

    const int E  = in_sizes[1];          // 800000 edges
    const int n4 = out_size / 4;         // out_size = 50000*64 floats

    spmm_zero_kernel<<<(n4 + THREADS - 1) / THREADS, THREADS, 0, stream>>>(
        (float4*)out, n4);

    spmm_scatter_kernel<<<(E + EDGES_PER_BLOCK - 1) / EDGES_PER_BLOCK, THREADS, 0, stream>>>(
        (const float4*)x, rows, cols, vals, ws, idx, out, E);
}